// GMemN2N_79809082294946
// MI455X (gfx1250) — compile-verified
//
#include <hip/hip_runtime.h>
#include <hip/hip_bf16.h>
#include <math.h>

#define VOC   50257
#define EMB   128
#define N_MEM 8192
#define T_Q   50
#define T_M   50
#define N_HOPS 3

typedef float v2f __attribute__((ext_vector_type(2)));
typedef float v8f __attribute__((ext_vector_type(8)));

// ---------------------------------------------------------------------------
// u[j] = sum_t Wb[query[t]][j]
// ---------------------------------------------------------------------------
__global__ void k_embed_query(const int* __restrict__ query,
                              const float* __restrict__ Wb,
                              float* __restrict__ u) {
    __shared__ int toks[T_Q];
    int j = threadIdx.x;
    if (j < T_Q) toks[j] = query[j];
    __syncthreads();
    float acc = 0.f;
    for (int t = 0; t < T_Q; ++t)
        acc += Wb[(long)toks[t] * EMB + j];
    u[j] = acc;
}

// ---------------------------------------------------------------------------
// mem_in[m][j] = sum_t Wa[story[m][t]][j] + TA[m][j]   (same for out/Wc/TC)
// one block per memory, 128 threads (one per embedding column)
// ---------------------------------------------------------------------------
__global__ void k_embed_story(const int* __restrict__ story,
                              const float* __restrict__ Wa,
                              const float* __restrict__ Wc,
                              const float* __restrict__ TA,
                              const float* __restrict__ TC,
                              float* __restrict__ mem_in,
                              float* __restrict__ mem_out) {
    __shared__ int toks[T_M];
    int m = blockIdx.x;
    int j = threadIdx.x;
    if (j < T_M) toks[j] = story[m * T_M + j];
    __syncthreads();
    float si = TA[m * EMB + j];
    float so = TC[m * EMB + j];
    for (int t = 0; t < T_M; ++t) {
        long r = (long)toks[t] * EMB + j;
        si += Wa[r];
        so += Wc[r];
    }
    mem_in [m * EMB + j] = si;
    mem_out[m * EMB + j] = so;
}

// ---------------------------------------------------------------------------
// out[n] = dot(u[0:128], rows[n][0:128])  via V_WMMA_F32_16X16X4_F32
// A (16x4) = u broadcast down rows; B (4x16) = 16-row tile of rows^T.
// One wave handles 16 output rows; 32 chained WMMAs cover K=128.
// ---------------------------------------------------------------------------
__global__ void k_rowdot_wmma(const float* __restrict__ rows,
                              const float* __restrict__ u,
                              float* __restrict__ out,
                              int nrows) {
    __shared__ float su[EMB];
    int tid = threadIdx.x;
    if (tid < EMB) su[tid] = u[tid];
    __syncthreads();

    int lane = tid & 31;
    int wave = tid >> 5;
    int tile = blockIdx.x * (blockDim.x >> 5) + wave;   // 16 rows per tile
    long base = (long)tile * 16;

    int n    = lane & 15;     // output column within tile
    int half = lane >> 4;     // K-pair select (0 -> K=0,1 ; 1 -> K=2,3)

    long row = base + n;
    if (row >= nrows) row = nrows - 1;                  // clamp loads, EXEC stays full
    const float* rp = rows + row * EMB + 2 * half;

    v8f acc = {};
#pragma unroll
    for (int k0 = 0; k0 < EMB; k0 += 4) {
        // A layout (documented): lanes 0-15 carry K=k0,k0+1; lanes 16-31 K=k0+2,k0+3
        v2f a = *(const v2f*)&su[k0 + 2 * half];
        // B layout (mirrors A's K split): lane n holds B[2*half..+1][n] = rows[row][k0+2*half..]
        v2f b = *(const v2f*)(rp + k0);
        acc = __builtin_amdgcn_wmma_f32_16x16x4_f32(
            /*neg_a=*/false, a, /*neg_b=*/false, b,
            /*c_mod=*/(short)0, acc, /*reuse_a=*/false, /*reuse_b=*/false);
    }
    // Every D row is identical; lane L holds D[8*half][L%16] in VGPR0.
    if (half == 0 && base + n < nrows)
        out[base + n] = acc[0];
}

// ---------------------------------------------------------------------------
// in-place: scores -> exp(scores - max); Z = sum(exp). Single block.
// ---------------------------------------------------------------------------
__global__ void k_softmax(float* __restrict__ scores, float* __restrict__ Z) {
    __shared__ float red[1024];
    int tid = threadIdx.x;
    float mx = -INFINITY;
    for (int i = tid; i < N_MEM; i += 1024) mx = fmaxf(mx, scores[i]);
    red[tid] = mx; __syncthreads();
    for (int s = 512; s > 0; s >>= 1) {
        if (tid < s) red[tid] = fmaxf(red[tid], red[tid + s]);
        __syncthreads();
    }
    float gmax = red[0];
    __syncthreads();
    float sum = 0.f;
    for (int i = tid; i < N_MEM; i += 1024) {
        float e = expf(scores[i] - gmax);
        scores[i] = e;
        sum += e;
    }
    red[tid] = sum; __syncthreads();
    for (int s = 512; s > 0; s >>= 1) {
        if (tid < s) red[tid] += red[tid + s];
        __syncthreads();
    }
    if (tid == 0) *Z = red[0];
}

// ---------------------------------------------------------------------------
// partial[b][j] = sum_{m in chunk b} e[m] * mem_out[m][j]
// ---------------------------------------------------------------------------
#define WGT_BLOCKS 64
__global__ void k_weighted(const float* __restrict__ e,
                           const float* __restrict__ mem_out,
                           float* __restrict__ partial) {
    int j = threadIdx.x;
    int b = blockIdx.x;
    const int chunk = N_MEM / WGT_BLOCKS;
    int m0 = b * chunk;
    float acc = 0.f;
    for (int m = m0; m < m0 + chunk; ++m)
        acc += e[m] * mem_out[m * EMB + j];
    partial[b * EMB + j] = acc;
}

// ---------------------------------------------------------------------------
// w = (sum partials)/Z ; t = sigmoid(Wt_w @ u + Wt_b) ;
// u = u*(1-t) + (H_w @ w + H_b)*t.   Single block, 128 threads.
// ---------------------------------------------------------------------------
__global__ void k_update(const float* __restrict__ partial,
                         const float* __restrict__ Z,
                         const float* __restrict__ Wt_w,
                         const float* __restrict__ Wt_b,
                         const float* __restrict__ H_w,
                         const float* __restrict__ H_b,
                         float* __restrict__ u) {
    __shared__ float su[EMB], sw[EMB];
    int j = threadIdx.x;
    su[j] = u[j];
    float acc = 0.f;
    for (int b = 0; b < WGT_BLOCKS; ++b) acc += partial[b * EMB + j];
    sw[j] = acc / (*Z);
    __syncthreads();
    float tdot = Wt_b[j];
    float hdot = H_b[j];
    for (int k = 0; k < EMB; ++k) {
        tdot += Wt_w[j * EMB + k] * su[k];
        hdot += H_w [j * EMB + k] * sw[k];
    }
    float t = 1.f / (1.f + expf(-tdot));
    u[j] = su[j] * (1.f - t) + hdot * t;
}

// ---------------------------------------------------------------------------
// stats[0] = max(logits); stats[1] = log(sum exp(logits - max)). Single block.
// ---------------------------------------------------------------------------
__global__ void k_lsm_reduce(const float* __restrict__ logits,
                             float* __restrict__ stats) {
    __shared__ float red[1024];
    int tid = threadIdx.x;
    float mx = -INFINITY;
    for (int i = tid; i < VOC; i += 1024) mx = fmaxf(mx, logits[i]);
    red[tid] = mx; __syncthreads();
    for (int s = 512; s > 0; s >>= 1) {
        if (tid < s) red[tid] = fmaxf(red[tid], red[tid + s]);
        __syncthreads();
    }
    float gmax = red[0];
    __syncthreads();
    float sum = 0.f;
    for (int i = tid; i < VOC; i += 1024) sum += expf(logits[i] - gmax);
    red[tid] = sum; __syncthreads();
    for (int s = 512; s > 0; s >>= 1) {
        if (tid < s) red[tid] += red[tid + s];
        __syncthreads();
    }
    if (tid == 0) { stats[0] = gmax; stats[1] = logf(red[0]); }
}

__global__ void k_lsm_write(const float* __restrict__ logits,
                            const float* __restrict__ stats,
                            float* __restrict__ out) {
    int v = blockIdx.x * blockDim.x + threadIdx.x;
    if (v < VOC) out[v] = logits[v] - stats[0] - stats[1];
}

// ---------------------------------------------------------------------------
extern "C" void kernel_launch(void* const* d_in, const int* in_sizes, int n_in,
                              void* d_out, int out_size, void* d_ws, size_t ws_size,
                              hipStream_t stream) {
    (void)in_sizes; (void)n_in; (void)out_size; (void)ws_size;

    const int*   query = (const int*)  d_in[0];
    const int*   story = (const int*)  d_in[1];
    const float* Wa    = (const float*)d_in[2];
    const float* Wc    = (const float*)d_in[3];
    const float* Wb    = (const float*)d_in[4];
    const float* Wt_w  = (const float*)d_in[5];
    const float* Wt_b  = (const float*)d_in[6];
    const float* H_w   = (const float*)d_in[7];
    const float* H_b   = (const float*)d_in[8];
    const float* Wout  = (const float*)d_in[9];
    const float* TA    = (const float*)d_in[10];
    const float* TC    = (const float*)d_in[11];
    float*       out   = (float*)d_out;

    // workspace layout (floats)
    float* ws = (float*)d_ws;
    float* u        = ws;                         // 128
    float* mem_in   = u       + 128;              // 8192*128
    float* mem_out  = mem_in  + N_MEM * EMB;      // 8192*128
    float* scores   = mem_out + N_MEM * EMB;      // 8192
    float* partial  = scores  + N_MEM;            // 64*128
    float* Z        = partial + WGT_BLOCKS * EMB; // 1
    float* logits   = Z + 128;                    // 50257 (padded start)
    float* stats    = logits + ((VOC + 127) & ~127); // 2

    // 1) embeddings
    k_embed_query<<<1, EMB, 0, stream>>>(query, Wb, u);
    k_embed_story<<<N_MEM, EMB, 0, stream>>>(story, Wa, Wc, TA, TC, mem_in, mem_out);

    // 2) hops
    const int score_tiles  = N_MEM / 16;                    // 512
    const int score_blocks = score_tiles / 8;               // 64 (8 waves/block)
    for (int hop = 0; hop < N_HOPS; ++hop) {
        k_rowdot_wmma<<<score_blocks, 256, 0, stream>>>(mem_in, u, scores, N_MEM);
        k_softmax<<<1, 1024, 0, stream>>>(scores, Z);
        k_weighted<<<WGT_BLOCKS, EMB, 0, stream>>>(scores, mem_out, partial);
        k_update<<<1, EMB, 0, stream>>>(partial, Z, Wt_w, Wt_b, H_w, H_b, u);
    }

    // 3) output projection + log-softmax
    const int logit_tiles  = (VOC + 15) / 16;               // 3142
    const int logit_blocks = (logit_tiles + 7) / 8;         // 393
    k_rowdot_wmma<<<logit_blocks, 256, 0, stream>>>(Wout, u, logits, VOC);
    k_lsm_reduce<<<1, 1024, 0, stream>>>(logits, stats);
    k_lsm_write<<<(VOC + 255) / 256, 256, 0, stream>>>(logits, stats, out);
}